// PatchQuantumGenerator_86741159510221
// MI455X (gfx1250) — compile-verified
//
#include <hip/hip_runtime.h>
#include <math.h>

// ---------------------------------------------------------------------------
// PatchQuantumGenerator on MI455X (gfx1250):
//   out[b, g*16+j] = |(U_g v_b)[j]|^2 / max_j' |(U_g v_b)[j']|^2   (j,j' < 16)
// where v_b is the REAL product state after the RY(noise) layer and U_g is a
// fixed 32x32 complex unitary per generator. Cast as a real fp32 GEMM
// (M=65536, K=32, N=128) on V_WMMA_F32_16X16X4_F32, plus a tiny epilogue.
// ---------------------------------------------------------------------------

typedef float v2f __attribute__((ext_vector_type(2)));
typedef float v8f __attribute__((ext_vector_type(8)));

#define NQ    5
#define DIM   32           // 2^NQ state size
#define NGEN  4
#define NLAY  4
#define NCOLS 128          // 4 gen * {re,im} * 16 outputs
#define WPB   4            // waves per block
#define ROWS_PER_WAVE 16

// ---------------------------------------------------------------------------
// Kernel 1: build B[k][n] = {Re,Im} of U_g[j][k] for j<16.
// Thread (g,k) simulates circuit column k entirely in registers
// (all gate loops fully unrolled -> constant indexing, no scratch).
// ---------------------------------------------------------------------------
__global__ void build_unitaries(const float* __restrict__ w,
                                float* __restrict__ Bmat) {
  const int tid = threadIdx.x;
  if (tid >= NGEN * DIM) return;
  const int g = tid >> 5;
  const int k = tid & 31;

  float sr[DIM], si[DIM];
#pragma unroll
  for (int i = 0; i < DIM; ++i) { sr[i] = (i == k) ? 1.0f : 0.0f; si[i] = 0.0f; }

#pragma unroll
  for (int l = 0; l < NLAY; ++l) {
    const int r = (l % (NQ - 1)) + 1;   // StronglyEntanglingLayers ranges
    // --- Rot(phi, theta, omega) on each qubit ---
#pragma unroll
    for (int q = 0; q < NQ; ++q) {
      const float* wq = w + (((g * NLAY + l) * NQ + q) * 3);
      const float phi = wq[0], th = wq[1], om = wq[2];
      float ct, st, ca, sa, cb, cbs;
      sincosf(0.5f * th, &st, &ct);
      sincosf(0.5f * (phi + om), &sa, &ca);
      sincosf(0.5f * (phi - om), &cbs, &cb);   // cbs = sin(b), cb = cos(b)
      // U00 = e^{-ia} c, U01 = -e^{+ib} s, U10 = e^{-ib} s, U11 = e^{+ia} c
      const float u00r =  ca * ct,  u00i = -sa * ct;
      const float u01r = -cb * st,  u01i = -cbs * st;
      const float u10r =  cb * st,  u10i = -cbs * st;
      const float u11r =  ca * ct,  u11i =  sa * ct;
      const int mask = 1 << (4 - q);
#pragma unroll
      for (int i = 0; i < DIM; ++i) {
        if ((i & mask) == 0) {
          const int j = i | mask;
          const float ar = sr[i], ai = si[i], br = sr[j], bi = si[j];
          sr[i] = u00r * ar - u00i * ai + u01r * br - u01i * bi;
          si[i] = u00r * ai + u00i * ar + u01r * bi + u01i * br;
          sr[j] = u10r * ar - u10i * ai + u11r * br - u11i * bi;
          si[j] = u10r * ai + u10i * ar + u11r * bi + u11i * br;
        }
      }
    }
    // --- CNOT ring with range r (sequential, order matters) ---
#pragma unroll
    for (int q = 0; q < NQ; ++q) {
      const int c = q, t = (q + r) % NQ;
      const int mc = 1 << (4 - c), mt = 1 << (4 - t);
#pragma unroll
      for (int i = 0; i < DIM; ++i) {
        if ((i & mc) != 0 && (i & mt) == 0) {
          const int j = i | mt;
          const float tr = sr[i], ti = si[i];
          sr[i] = sr[j]; si[i] = si[j];
          sr[j] = tr;    si[j] = ti;
        }
      }
    }
  }
  // Store only rows j<16 (ancilla==0 patch): columns n = g*32 + comp*16 + j
#pragma unroll
  for (int j = 0; j < 16; ++j) {
    Bmat[k * NCOLS + g * 32 + j]      = sr[j];
    Bmat[k * NCOLS + g * 32 + 16 + j] = si[j];
  }
}

// ---------------------------------------------------------------------------
// Kernel 2: fp32 WMMA GEMM + epilogue. One wave per 16 batch rows.
//   A (16x4 f32):  VGPR0 = K (0|2), VGPR1 = K (1|3); M = lane&15, half by lane>=16
//   B (4x16 f32):  lane&15 = N, lane-half selects K pair (mirror of A)
//   C/D (16x16):   VGPR r -> M = r + 8*(lane>=16), N = lane&15
// ---------------------------------------------------------------------------
__global__ __launch_bounds__(WPB * 32)
void qgen_wmma(const float* __restrict__ x,
               const float* __restrict__ Bmat,
               float* __restrict__ out) {
  __shared__ float lds[WPB][8][16][16];   // [wave][tile][M][N] = 32 KB

  const int  lane = threadIdx.x & 31;
  const int  wave = threadIdx.x >> 5;
  const bool hiHalf = (lane >= 16);
  const int  m    = lane & 15;
  const int  b0   = (blockIdx.x * WPB + wave) * ROWS_PER_WAVE;
  const int  row  = b0 + m;

  // --- per-qubit cos/sin(x/2), then the 32 product-state amplitudes ---
  float cq[NQ], sq[NQ];
#pragma unroll
  for (int q = 0; q < NQ; ++q) {
    sincosf(0.5f * x[row * NQ + q], &sq[q], &cq[q]);
  }
  const float t01[4] = {cq[0]*cq[1], cq[0]*sq[1], sq[0]*cq[1], sq[0]*sq[1]};
  const float t34[4] = {cq[3]*cq[4], cq[3]*sq[4], sq[3]*cq[4], sq[3]*sq[4]};
  float v[DIM];
#pragma unroll
  for (int i = 0; i < DIM; ++i) {
    const float f2 = ((i >> 2) & 1) ? sq[2] : cq[2];
    v[i] = t01[i >> 3] * f2 * t34[i & 3];   // bit_q(i) = (i >> (4-q)) & 1
  }

  // --- A fragments: afrag[kc] covers K = 4*kc .. 4*kc+3 ---
  v2f afrag[8];
#pragma unroll
  for (int kc = 0; kc < 8; ++kc) {
    afrag[kc].x = hiHalf ? v[kc * 4 + 2] : v[kc * 4 + 0];
    afrag[kc].y = hiHalf ? v[kc * 4 + 3] : v[kc * 4 + 1];
  }

  const int kbase = hiHalf ? 2 : 0;
  const int mbase = hiHalf ? 8 : 0;

  // --- 8 N-tiles x 8 K-steps of v_wmma_f32_16x16x4_f32 ---
#pragma unroll
  for (int nt = 0; nt < 8; ++nt) {
    const int n = nt * 16 + m;
    v8f acc = {0.f, 0.f, 0.f, 0.f, 0.f, 0.f, 0.f, 0.f};
#pragma unroll
    for (int kc = 0; kc < 8; ++kc) {
      v2f bf;
      bf.x = Bmat[(kc * 4 + kbase)     * NCOLS + n];
      bf.y = Bmat[(kc * 4 + kbase + 1) * NCOLS + n];
      acc = __builtin_amdgcn_wmma_f32_16x16x4_f32(
          false, afrag[kc], false, bf, (short)0, acc, false, false);
    }
#pragma unroll
    for (int r = 0; r < 8; ++r) {
      lds[wave][nt][mbase + r][m] = acc[r];
    }
  }

  __syncthreads();   // cross-lane LDS visibility (DScnt wait + barrier)

  // --- epilogue: 64 (row,g) tasks per wave; tile 2g = Re, 2g+1 = Im ---
#pragma unroll
  for (int t = 0; t < 2; ++t) {
    const int g = (lane >> 4) + 2 * t;
    float p[16];
    float mx = 0.0f;
#pragma unroll
    for (int j = 0; j < 16; ++j) {
      const float fr = lds[wave][2 * g + 0][m][j];
      const float fi = lds[wave][2 * g + 1][m][j];
      p[j] = fr * fr + fi * fi;
      mx = fmaxf(mx, p[j]);
    }
    const float inv = 1.0f / mx;   // sum normalization cancels in p0/max(p0)
    float4* o4 = reinterpret_cast<float4*>(out + (size_t)row * 64 + g * 16);
#pragma unroll
    for (int jj = 0; jj < 4; ++jj) {
      o4[jj] = make_float4(p[jj*4+0]*inv, p[jj*4+1]*inv,
                           p[jj*4+2]*inv, p[jj*4+3]*inv);
    }
  }
}

// ---------------------------------------------------------------------------
extern "C" void kernel_launch(void* const* d_in, const int* in_sizes, int n_in,
                              void* d_out, int out_size, void* d_ws, size_t ws_size,
                              hipStream_t stream) {
  const float* x = (const float*)d_in[0];        // [B, 5]
  const float* w = (const float*)d_in[1];        // [4, 4, 5, 3]
  float* out  = (float*)d_out;                   // [B, 64]
  float* Bmat = (float*)d_ws;                    // 32*128 f32 = 16 KB scratch

  const int B = in_sizes[0] / NQ;

  build_unitaries<<<1, NGEN * DIM, 0, stream>>>(w, Bmat);

  const int rows_per_block = WPB * ROWS_PER_WAVE;          // 64
  qgen_wmma<<<B / rows_per_block, WPB * 32, 0, stream>>>(x, Bmat, out);
}